// iSogCLR_Plus_Loss_90632399880306
// MI455X (gfx1250) — compile-verified
//
#include <hip/hip_runtime.h>
#include <math.h>

typedef __attribute__((ext_vector_type(2))) float v2f;
typedef __attribute__((ext_vector_type(8))) float v8f;

#define GAMMA_S   0.8f
#define GAMMA_U   0.8f
#define RHO_I     0.1f
#define RHO_T     0.1f
#define EPS_C     1e-10f
#define TAU_MIN_C 0.005f
#define TAU_MAX_C 1.0f
#define GRAD_CLIP_C 5.0f
#define ETA_INIT_C 0.01f
#define ETA_MIN_C  0.0001f
#define PI_C 3.14159265358979323846f

// ---------------------------------------------------------------------------
// Kernel 1: fp32 WMMA GEMM  sim = Img (BxD) * Txt^T (BxD)^T, row-major output
// Block = 128 threads (4 waves), block tile 128x128, wave tile 64x64.
// Uses V_WMMA_F32_16X16X4_F32; A/B fragments are 2 VGPRs (float2) per lane.
// 32-bit A 16x4 layout: lane<16 -> M=lane, K={k0,k0+1}; lane>=16 -> K={k0+2,k0+3}.
// ---------------------------------------------------------------------------
__global__ __launch_bounds__(128) void sim_gemm_wmma_f32(
    const float* __restrict__ A, const float* __restrict__ Bm,
    float* __restrict__ C, int Bsz, int D)
{
    const int lane = threadIdx.x & 31;
    const int wave = threadIdx.x >> 5;            // 0..3
    const int m0 = blockIdx.y * 128 + (wave & 1) * 64;
    const int n0 = blockIdx.x * 128 + (wave >> 1) * 64;
    const int lr = lane & 15;                     // row within 16
    const int kh = lane >> 4;                     // 0/1 -> K pair offset 2*kh

    const float* aBase = A  + (size_t)(m0 + lr) * D + 2 * kh;
    const float* bBase = Bm + (size_t)(n0 + lr) * D + 2 * kh;

    v8f acc[4][4] = {};

    for (int k = 0; k < D; k += 4) {
        v2f af[4], bf[4];
#pragma unroll
        for (int i = 0; i < 4; ++i) {
            af[i] = *(const v2f*)(aBase + (size_t)(i * 16) * D + k);
            bf[i] = *(const v2f*)(bBase + (size_t)(i * 16) * D + k);
        }
#pragma unroll
        for (int mi = 0; mi < 4; ++mi)
#pragma unroll
            for (int ni = 0; ni < 4; ++ni)
                acc[mi][ni] = __builtin_amdgcn_wmma_f32_16x16x4_f32(
                    false, af[mi], false, bf[ni],
                    (short)0, acc[mi][ni], false, false);
    }

    // C/D 16x16 fp32 layout: VGPR r: lanes 0-15 -> M=r, lanes 16-31 -> M=8+r; N=lane&15
#pragma unroll
    for (int mi = 0; mi < 4; ++mi)
#pragma unroll
        for (int ni = 0; ni < 4; ++ni)
#pragma unroll
            for (int r = 0; r < 8; ++r) {
                int row = m0 + mi * 16 + r + 8 * kh;
                int col = n0 + ni * 16 + lr;
                C[(size_t)row * Bsz + col] = acc[mi][ni][r];
            }
}

// ---------------------------------------------------------------------------
// Online (max, sum(exp), sum(exp*x)) merge
// ---------------------------------------------------------------------------
__device__ __forceinline__ void olmerge(float& m, float& se, float& sei,
                                        float m2, float se2, float sei2)
{
    float M  = fmaxf(m, m2);
    float c1 = __expf(m - M);
    float c2 = __expf(m2 - M);
    se  = se * c1 + se2 * c2;
    sei = sei * c1 + sei2 * c2;
    m = M;
}

__device__ __forceinline__ float eta_cur_dev(int ep, int mep)
{
    return ETA_MIN_C + (ETA_INIT_C - ETA_MIN_C) *
           __cosf(PI_C * ((float)ep / (float)mep) * 0.5f);
}

// ---------------------------------------------------------------------------
// Kernel 2: per-row (image side). One block of 256 threads per row.
// ---------------------------------------------------------------------------
__global__ __launch_bounds__(256) void row_reduce_image(
    const float* __restrict__ sim,
    const float* __restrict__ tau_I, const float* __restrict__ b_I,
    const float* __restrict__ s_I,   const float* __restrict__ u_I,
    const int* __restrict__ image_ids,
    const int* __restrict__ epoch_p, const int* __restrict__ max_epoch_p,
    float* __restrict__ out, float* __restrict__ imgLoss,
    int Bsz, long long Nst)
{
    const int i   = blockIdx.x;
    const int tid = threadIdx.x;
    const int id  = image_ids[i];
    const float* row = sim + (size_t)i * Bsz;
    const float diag = row[i];
    const float tau  = tau_I[id];
    const float itau = 1.0f / tau;

    float m = -__builtin_huge_valf(), se = 0.0f, sei = 0.0f;
    for (int j = tid; j < Bsz; j += 256) {
        float x = (row[j] - diag) * itau;
        if (x > m) {
            float c = __expf(m - x);
            se = se * c + 1.0f;
            sei = sei * c + x;
            m = x;
        } else {
            float e = __expf(x - m);
            se += e;
            sei += e * x;
        }
    }
    // wave32 reduction
    for (int off = 16; off; off >>= 1) {
        float m2   = __shfl_xor(m,   off, 32);
        float se2  = __shfl_xor(se,  off, 32);
        float sei2 = __shfl_xor(sei, off, 32);
        olmerge(m, se, sei, m2, se2, sei2);
    }
    __shared__ float sm[8], sse[8], ssei[8];
    if ((tid & 31) == 0) { sm[tid >> 5] = m; sse[tid >> 5] = se; ssei[tid >> 5] = sei; }
    __syncthreads();
    if (tid == 0) {
        m = sm[0]; se = sse[0]; sei = ssei[0];
        for (int w = 1; w < 8; ++w) olmerge(m, se, sei, sm[w], sse[w], ssei[w]);

        float old_b = b_I[id];
        float new_b = fmaxf(m, old_b);
        float scale = __expf(m - new_b);
        se  *= scale;
        sei *= scale;

        float inv_bm1 = 1.0f / (float)(Bsz - 1);
        float g = se * inv_bm1;
        float s_new = (1.0f - GAMMA_S) * s_I[id] * __expf(old_b - new_b) + GAMMA_S * g;
        float sum_w_idt = sei / (s_new + EPS_C) * inv_bm1;
        float loss_i = tau * sum_w_idt;
        float grad_tau = __logf(s_new) + new_b + RHO_I - sum_w_idt;
        float gclip = fminf(fmaxf(grad_tau, -GRAD_CLIP_C), GRAD_CLIP_C);
        float u_new = (1.0f - GAMMA_U) * u_I[id] + GAMMA_U * gclip;
        float eta = eta_cur_dev(epoch_p[0], max_epoch_p[0]);
        float tau_new = fminf(fmaxf(tau - eta * u_new, TAU_MIN_C), TAU_MAX_C);

        out[i] = g;                            // g_I
        out[2 * (size_t)Bsz + i] = grad_tau;   // grad_tau_image
        imgLoss[i] = loss_i;

        float* base = out + 4 * (size_t)Bsz + 3;
        base[0 * Nst + id] = s_new;            // s_I_out
        base[2 * Nst + id] = u_new;            // u_I_out
        base[4 * Nst + id] = new_b;            // b_I_out
        base[6 * Nst + id] = tau_new;          // tau_I_out
    }
}

// ---------------------------------------------------------------------------
// Kernel 3: per-column (text side). 256 threads, 32 columns per block.
// lane = column (coalesced 128B reads per wave); 8 waves split the rows.
// ---------------------------------------------------------------------------
__global__ __launch_bounds__(256) void col_reduce_text(
    const float* __restrict__ sim,
    const float* __restrict__ tau_T, const float* __restrict__ b_T,
    const float* __restrict__ s_T,   const float* __restrict__ u_T,
    const int* __restrict__ text_ids,
    const int* __restrict__ epoch_p, const int* __restrict__ max_epoch_p,
    float* __restrict__ out, float* __restrict__ txtLoss,
    int Bsz, long long Nst)
{
    const int lane = threadIdx.x & 31;
    const int w    = threadIdx.x >> 5;
    const int c    = blockIdx.x * 32 + lane;
    const int id   = text_ids[c];
    const float diag = sim[(size_t)c * Bsz + c];
    const float tau  = tau_T[id];
    const float itau = 1.0f / tau;

    float m = -__builtin_huge_valf(), se = 0.0f, sei = 0.0f;
    for (int r = w; r < Bsz; r += 8) {
        float x = (sim[(size_t)r * Bsz + c] - diag) * itau;
        if (x > m) {
            float cc = __expf(m - x);
            se = se * cc + 1.0f;
            sei = sei * cc + x;
            m = x;
        } else {
            float e = __expf(x - m);
            se += e;
            sei += e * x;
        }
    }
    __shared__ float sm[8][32], sse[8][32], ssei[8][32];
    sm[w][lane] = m; sse[w][lane] = se; ssei[w][lane] = sei;
    __syncthreads();
    if (w == 0) {
        for (int ww = 1; ww < 8; ++ww)
            olmerge(m, se, sei, sm[ww][lane], sse[ww][lane], ssei[ww][lane]);

        float old_b = b_T[id];
        float new_b = fmaxf(m, old_b);
        float scale = __expf(m - new_b);
        se  *= scale;
        sei *= scale;

        float inv_bm1 = 1.0f / (float)(Bsz - 1);
        float g = se * inv_bm1;
        float s_new = (1.0f - GAMMA_S) * s_T[id] * __expf(old_b - new_b) + GAMMA_S * g;
        float sum_w_tdt = sei / (s_new + EPS_C) * inv_bm1;
        float loss_c = tau * sum_w_tdt;
        float grad_tau = __logf(s_new) + new_b + RHO_T - sum_w_tdt;
        float gclip = fminf(fmaxf(grad_tau, -GRAD_CLIP_C), GRAD_CLIP_C);
        float u_new = (1.0f - GAMMA_U) * u_T[id] + GAMMA_U * gclip;
        float eta = eta_cur_dev(epoch_p[0], max_epoch_p[0]);
        float tau_new = fminf(fmaxf(tau - eta * u_new, TAU_MIN_C), TAU_MAX_C);

        out[(size_t)Bsz + c] = g;                  // g_T
        out[3 * (size_t)Bsz + c] = grad_tau;       // grad_tau_text
        txtLoss[c] = loss_c;

        float* base = out + 4 * (size_t)Bsz + 3;
        base[1 * Nst + id] = s_new;                // s_T_out
        base[3 * Nst + id] = u_new;                // u_T_out
        base[5 * Nst + id] = new_b;                // b_T_out
        base[7 * Nst + id] = tau_new;              // tau_T_out
    }
}

// ---------------------------------------------------------------------------
// Kernel 0: copy the 8 state arrays input -> output (before scatters)
// ---------------------------------------------------------------------------
__global__ __launch_bounds__(256) void copy_state(
    const float* __restrict__ s_I, const float* __restrict__ s_T,
    const float* __restrict__ u_I, const float* __restrict__ u_T,
    const float* __restrict__ b_I, const float* __restrict__ b_T,
    const float* __restrict__ tau_I, const float* __restrict__ tau_T,
    float* __restrict__ base, long long N)
{
    long long i = (long long)blockIdx.x * blockDim.x + threadIdx.x;
    if (i < N) {
        base[0 * N + i] = s_I[i];
        base[1 * N + i] = s_T[i];
        base[2 * N + i] = u_I[i];
        base[3 * N + i] = u_T[i];
        base[4 * N + i] = b_I[i];
        base[5 * N + i] = b_T[i];
        base[6 * N + i] = tau_I[i];
        base[7 * N + i] = tau_T[i];
    }
}

// ---------------------------------------------------------------------------
// Kernel 4: scalars (total_loss, avg taus). Single block, deterministic.
// ---------------------------------------------------------------------------
__global__ __launch_bounds__(256) void finalize_scalars(
    const float* __restrict__ imgLoss, const float* __restrict__ txtLoss,
    const float* __restrict__ tau_I, const float* __restrict__ tau_T,
    const int* __restrict__ image_ids, const int* __restrict__ text_ids,
    float* __restrict__ out, int Bsz)
{
    __shared__ float sa[256], sb[256], sc[256], sd[256];
    float a = 0.f, b = 0.f, c = 0.f, d = 0.f;
    for (int i = threadIdx.x; i < Bsz; i += 256) {
        a += imgLoss[i];
        b += txtLoss[i];
        c += tau_I[image_ids[i]];
        d += tau_T[text_ids[i]];
    }
    sa[threadIdx.x] = a; sb[threadIdx.x] = b; sc[threadIdx.x] = c; sd[threadIdx.x] = d;
    __syncthreads();
    for (int off = 128; off; off >>= 1) {
        if (threadIdx.x < off) {
            sa[threadIdx.x] += sa[threadIdx.x + off];
            sb[threadIdx.x] += sb[threadIdx.x + off];
            sc[threadIdx.x] += sc[threadIdx.x + off];
            sd[threadIdx.x] += sd[threadIdx.x + off];
        }
        __syncthreads();
    }
    if (threadIdx.x == 0) {
        float invB = 1.0f / (float)Bsz;
        out[4 * (size_t)Bsz + 0] = sa[0] * invB + sb[0] * invB;  // total_loss
        out[4 * (size_t)Bsz + 1] = sc[0] * invB;                 // avg_image_tau
        out[4 * (size_t)Bsz + 2] = sd[0] * invB;                 // avg_text_tau
    }
}

// ---------------------------------------------------------------------------
extern "C" void kernel_launch(void* const* d_in, const int* in_sizes, int n_in,
                              void* d_out, int out_size, void* d_ws, size_t ws_size,
                              hipStream_t stream)
{
    const float* imf   = (const float*)d_in[0];
    const float* txf   = (const float*)d_in[1];
    const float* s_I   = (const float*)d_in[2];
    const float* s_T   = (const float*)d_in[3];
    const float* tau_I = (const float*)d_in[4];
    const float* tau_T = (const float*)d_in[5];
    const float* u_I   = (const float*)d_in[6];
    const float* u_T   = (const float*)d_in[7];
    const float* b_I   = (const float*)d_in[8];
    const float* b_T   = (const float*)d_in[9];
    const int* image_ids = (const int*)d_in[10];
    const int* text_ids  = (const int*)d_in[11];
    const int* epoch_p   = (const int*)d_in[12];
    const int* max_ep_p  = (const int*)d_in[13];

    const int Bsz = in_sizes[10];            // 4096
    const int D   = in_sizes[0] / Bsz;       // 1024
    const long long Nst = in_sizes[2];       // 2.9M

    float* out = (float*)d_out;
    float* sim = (float*)d_ws;
    float* imgLoss = sim + (size_t)Bsz * Bsz;
    float* txtLoss = imgLoss + Bsz;

    float* stateBase = out + 4 * (size_t)Bsz + 3;

    // 0) copy state (scatters land on top afterwards)
    {
        long long blocks = (Nst + 255) / 256;
        copy_state<<<(int)blocks, 256, 0, stream>>>(s_I, s_T, u_I, u_T,
                                                    b_I, b_T, tau_I, tau_T,
                                                    stateBase, Nst);
    }
    // 1) similarity GEMM via fp32 WMMA
    {
        dim3 grid(Bsz / 128, Bsz / 128);
        sim_gemm_wmma_f32<<<grid, 128, 0, stream>>>(imf, txf, sim, Bsz, D);
    }
    // 2) image-side row reductions + scatter
    row_reduce_image<<<Bsz, 256, 0, stream>>>(sim, tau_I, b_I, s_I, u_I,
                                              image_ids, epoch_p, max_ep_p,
                                              out, imgLoss, Bsz, Nst);
    // 3) text-side column reductions + scatter
    col_reduce_text<<<Bsz / 32, 256, 0, stream>>>(sim, tau_T, b_T, s_T, u_T,
                                                  text_ids, epoch_p, max_ep_p,
                                                  out, txtLoss, Bsz, Nst);
    // 4) scalars
    finalize_scalars<<<1, 256, 0, stream>>>(imgLoss, txtLoss, tau_I, tau_T,
                                            image_ids, text_ids, out, Bsz);
}